// LocationDependentClassifier_39659728011726
// MI455X (gfx1250) — compile-verified
//
#include <hip/hip_runtime.h>
#include <hip/hip_bf16.h>
#include <math.h>

// CDNA5 (gfx1250) wave32 WMMA types
typedef __attribute__((ext_vector_type(2))) float v2f;   // 16x4 / 4x16 f32 fragment = 2 VGPRs
typedef __attribute__((ext_vector_type(8))) float v8f;   // 16x16 f32 C/D fragment = 8 VGPRs

#define CH_STRIDE   (512 * 512)
#define BATCH_STRIDE (3 * 512 * 512)
#define NUM_CLASSES 10

// One wave (32 threads) per block; each block reduces 16 batches via
// 48 chained V_WMMA_F32_16X16X4_F32 (A = 16 batches x 4 patch elems, B = ones).
__global__ __launch_bounds__(32)
void LocationDependentClassifier_39659728011726_kernel(const float* __restrict__ x,
                                                       float* __restrict__ out) {
    const int lane  = threadIdx.x;            // 0..31, wave32
    const int tile  = blockIdx.x;             // 0..3 -> batches [tile*16, tile*16+16)
    const int mrow  = lane & 15;              // M row (batch within tile) this lane feeds
    const int khalf = (lane >> 4) ? 2 : 0;    // lanes 16..31 supply K=2,3 (ISA A layout)

    const float* bp = x + (size_t)(tile * 16 + mrow) * BATCH_STRIDE;

    // ---- Phase 1: issue all 48 loads (2 consecutive floats each, 8B aligned) ----
    v2f a[48];
#pragma unroll
    for (int t = 0; t < 48; ++t) {
        const int e0 = 4 * t + khalf;         // element index in [0,192)
        const int c  = e0 >> 6;               // channel 0..2
        const int r  = (e0 >> 3) & 7;         // row within 8x8 patch
        const int j  = e0 & 7;                // col: 0,2,4,6 -> j and j+1 same row
        a[t] = *(const v2f*)(bp + c * CH_STRIDE + r * 512 + j);
    }

    // ---- Phase 2: 48 chained fp32 WMMAs: D = A(16x4) * ones(4x16) + C ----
    const v2f bones = {1.0f, 1.0f};           // all-ones B => D[m][n] = row-sum of A
    v8f acc = {};
#pragma unroll
    for (int t = 0; t < 48; ++t) {
        acc = __builtin_amdgcn_wmma_f32_16x16x4_f32(
            /*neg_a=*/false, a[t], /*neg_b=*/false, bones,
            /*c_mod=*/(short)0, acc, /*reuse_a=*/false, /*reuse_b=*/false);
    }

    // D layout (16x16 f32): lanes 0-15 hold M=0..7 in acc[0..7] (replicated over N);
    // lanes 16-31 hold M=8..15. Each lane col<10 writes one logit column for 8 batches.
    const int col   = lane & 15;
    const int bbase = tile * 16 + ((lane >> 4) ? 8 : 0);
    if (col < NUM_CLASSES) {
#pragma unroll
        for (int i = 0; i < 8; ++i) {
            const float mean = acc[i] / 192.0f;           // mean over 3*8*8
            const float t10  = truncf(mean * 10.0f);      // trunc toward zero (matches jnp.trunc)
            const int   ti   = (int)t10;
            const int   cls  = ((ti % NUM_CLASSES) + NUM_CLASSES) % NUM_CLASSES; // python mod
            out[(size_t)(bbase + i) * NUM_CLASSES + col] = (col == cls) ? 10.0f : 0.0f;
        }
    }
}

extern "C" void kernel_launch(void* const* d_in, const int* in_sizes, int n_in,
                              void* d_out, int out_size, void* d_ws, size_t ws_size,
                              hipStream_t stream) {
    const float* x = (const float*)d_in[0];   // (64, 3, 512, 512) fp32
    float* out     = (float*)d_out;           // (64, 10) fp32

    // 4 blocks x 1 wave; each wave owns 16 batches.
    LocationDependentClassifier_39659728011726_kernel<<<4, 32, 0, stream>>>(x, out);
}